// GausLJLayer_1288490189271
// MI455X (gfx1250) — compile-verified
//
#include <hip/hip_runtime.h>
#include <stdint.h>

#define N_LJ    3
#define N_GAUSS 4
#define NPARAM  21
#define TILE    256
#define NVEC4   ((TILE * NPARAM) / 4)   // 1344 float4s per tile
#define FULLROUNDS (NVEC4 / TILE)       // 5
#define TAILCNT    (NVEC4 % TILE)       // 64

#if __has_builtin(__builtin_amdgcn_global_load_async_to_lds_b128)
#define HAVE_ASYNC_LDS 1
#else
#define HAVE_ASYNC_LDS 0
#endif

typedef int v4i __attribute__((ext_vector_type(4)));

__global__ __launch_bounds__(TILE) void gauss_lj_kernel(
    const float* __restrict__ dist,
    const float* __restrict__ params,
    float* __restrict__ out,
    int n)
{
    __shared__ float tile[TILE * NPARAM];   // 256 rows * 21 f32 = 21 KB

    const int tid  = threadIdx.x;
    const int base = blockIdx.x * TILE;
    const int idx  = base + tid;
    const bool full_tile = (base + TILE) <= n;

    // Issue the per-thread distance load early so it overlaps the LDS DMA.
    float d = 0.0f;
    if (idx < n) d = dist[idx];

    float pp[NPARAM];

    if (full_tile) {
        // ---- Cooperative, fully-coalesced staging of the 21 KB param tile ----
        // Tile byte base = base*84, divisible by 16 -> float4-aligned.
        const float4* __restrict__ gsrc = (const float4*)(params + (size_t)base * NPARAM);
        float4* ldst = (float4*)tile;

#if HAVE_ASYNC_LDS
        // CDNA5 async DMA path: global -> LDS without a VGPR round trip.
        // 1344 = 5*256 + 64 -> straight-line issue, no loop mask management.
#pragma unroll
        for (int r = 0; r < FULLROUNDS; ++r) {
            const int i = tid + r * TILE;
            __builtin_amdgcn_global_load_async_to_lds_b128(
                (v4i*)(gsrc + i), (v4i*)(ldst + i), 0, 0);
        }
        if (tid < TAILCNT) {
            const int i = tid + FULLROUNDS * TILE;
            __builtin_amdgcn_global_load_async_to_lds_b128(
                (v4i*)(gsrc + i), (v4i*)(ldst + i), 0, 0);
        }
#if __has_builtin(__builtin_amdgcn_s_wait_asynccnt)
        __builtin_amdgcn_s_wait_asynccnt(0);
#else
        asm volatile("s_wait_asynccnt 0" ::: "memory");
#endif
#else
        // Fallback: plain b128 load -> ds_store (still fully coalesced).
        for (int i = tid; i < NVEC4; i += TILE) {
            ldst[i] = gsrc[i];
        }
#endif
        __syncthreads();

        // Row read from LDS: stride-21 (odd) across 32 lanes -> conflict-free.
        const float* __restrict__ row = &tile[tid * NPARAM];
#pragma unroll
        for (int k = 0; k < NPARAM; ++k) pp[k] = row[k];
    } else {
        // Ragged last block (not hit for B = 4M, kept for generality).
        if (idx < n) {
            const float* __restrict__ row = params + (size_t)idx * NPARAM;
#pragma unroll
            for (int k = 0; k < NPARAM; ++k) pp[k] = row[k];
        }
    }

    if (idx >= n) return;

    const float rd = 1.0f / d;

    float energy = 0.0f;
    float force  = 0.0f;

    // ---- Lennard-Jones terms ----
#pragma unroll
    for (int j = 0; j < N_LJ; ++j) {
        const float c     = pp[3 * j + 1];
        const float sigma = pp[3 * j + 2];
        const float ir    = sigma * rd;
        const float ir2   = ir * ir;
        const float ir6   = ir2 * ir2 * ir2;
        const float ir12  = ir6 * ir6;
        energy += 4.0f * c * (ir12 - ir6);
        // -4c(4 s^6/d^7 - 12 s^12/d^13) = 4c*rd*(12*ir12 - 4*ir6)
        force  += 4.0f * c * rd * (12.0f * ir12 - 4.0f * ir6);
    }

    // ---- Gaussian terms ----
#pragma unroll
    for (int j = 0; j < N_GAUSS; ++j) {
        const float A    = pp[3 * N_LJ + 3 * j + 0];
        const float mean = pp[3 * N_LJ + 3 * j + 1];
        const float sd   = pp[3 * N_LJ + 3 * j + 2];
        const float dm     = d - mean;
        const float inv_s2 = 1.0f / (sd * sd);
        const float dm2s   = dm * dm * inv_s2;
        const float e      = expf(-0.5f * dm2s);
        energy += A * e;
        // -(A*dm*e*dm^2/s^2)/s^2
        force  -= A * dm * e * dm2s * inv_s2;
    }

    out[idx]     = energy;  // energies
    out[n + idx] = force;   // forces (concatenated after energies)
}

extern "C" void kernel_launch(void* const* d_in, const int* in_sizes, int n_in,
                              void* d_out, int out_size, void* d_ws, size_t ws_size,
                              hipStream_t stream) {
    (void)n_in; (void)out_size; (void)d_ws; (void)ws_size;
    const float* dist   = (const float*)d_in[0];
    const float* params = (const float*)d_in[1];
    float* out          = (float*)d_out;
    const int n         = in_sizes[0];

    const int grid = (n + TILE - 1) / TILE;
    gauss_lj_kernel<<<grid, TILE, 0, stream>>>(dist, params, out, n);
}